// GCNMainBlock_71159018160288
// MI455X (gfx1250) — compile-verified
//
#include <hip/hip_runtime.h>

typedef float v2f __attribute__((ext_vector_type(2)));
typedef float v8f __attribute__((ext_vector_type(8)));

#define HID    128
#define KSTEPS 32   // 128 / K=4 per WMMA
#define NTILES 8    // 128 / 16 output cols per tile

// ---------------------------------------------------------------------------
// Kernel 1: h = x @ W^T + b via V_WMMA_F32_16X16X4_F32.
// One wave per 16-row M-tile. A-panel (16x128) preloaded in registers.
// A frag layout (32-bit 16x4): lanes 0-15 -> M=0..15, VGPR0=K0,VGPR1=K1;
// lanes 16-31 -> same M, K2/K3.  B frag mirrors with N instead of M.
// C/D layout: VGPR r, lane l -> row = r + 8*(l>>4), col = l&15.
// Writes h to workspace (for the edge gather) AND to out (self term).
// ---------------------------------------------------------------------------
__global__ __launch_bounds__(128)
void gcn_gemm_wmma(const float* __restrict__ x,
                   const float* __restrict__ W,
                   const float* __restrict__ b,
                   float* __restrict__ h,
                   float* __restrict__ out,
                   int n_nodes)
{
    const int wave   = blockIdx.x * (blockDim.x >> 5) + (threadIdx.x >> 5);
    const int lane   = threadIdx.x & 31;
    const int m_base = wave * 16;
    if (m_base >= n_nodes) return;   // whole wave exits together (EXEC all-1s for WMMA)

    const int rown  = lane & 15;          // M row (A) / N col (B) within tile
    const int khalf = (lane >> 4) << 1;   // 0 or 2: K offset within the 4-wide K step

    // Preload A fragments: 32 k-steps, one float2 per lane each.
    v2f a[KSTEPS];
    {
        const float* xrow = x + (size_t)(m_base + rown) * HID + khalf;
        #pragma unroll
        for (int k = 0; k < KSTEPS; ++k)
            a[k] = *reinterpret_cast<const v2f*>(xrow + 4 * k);
    }

    #pragma unroll
    for (int nt = 0; nt < NTILES; ++nt) {
        const int n_base = nt * 16;
        const float bn = b[n_base + rown];
        v8f c;
        #pragma unroll
        for (int i = 0; i < 8; ++i) c[i] = bn;   // C init = bias broadcast per column

        const float* wrow = W + (size_t)(n_base + rown) * HID + khalf;  // B[k][n] = W[n][k]
        #pragma unroll
        for (int k = 0; k < KSTEPS; ++k) {
            v2f bf = *reinterpret_cast<const v2f*>(wrow + 4 * k);
            c = __builtin_amdgcn_wmma_f32_16x16x4_f32(
                    /*neg_a=*/false, a[k], /*neg_b=*/false, bf,
                    /*c_mod=*/(short)0, c, /*reuse_a=*/false, /*reuse_b=*/false);
        }

        const int col   = n_base + rown;
        const int rbase = m_base + ((lane >> 4) << 3);
        #pragma unroll
        for (int r = 0; r < 8; ++r) {
            const size_t off = (size_t)(rbase + r) * HID + col;
            h[off]   = c[r];
            out[off] = c[r];
        }
    }
}

// ---------------------------------------------------------------------------
// Kernel 2: zero the degree array (graph-capture-safe init).
// ---------------------------------------------------------------------------
__global__ void gcn_zero_deg(int* __restrict__ deg, int n)
{
    int i = blockIdx.x * blockDim.x + threadIdx.x;
    if (i < n) deg[i] = 0;
}

// ---------------------------------------------------------------------------
// Kernel 3: scatter-sum. One wave (32 lanes) per edge; each lane moves a
// float4 (h is L2-resident: 25.6 MB << 192 MB) and issues 4 hardware
// global_atomic_add_f32 into out[dst]. Lane 0 counts the in-degree.
// ---------------------------------------------------------------------------
__global__ __launch_bounds__(256)
void gcn_scatter(const int* __restrict__ edge_index,
                 const float* __restrict__ h,
                 float* __restrict__ out,
                 int* __restrict__ deg,
                 int n_edges)
{
    const int tid = blockIdx.x * blockDim.x + threadIdx.x;
    const int e   = tid >> 5;
    if (e >= n_edges) return;
    const int lane = tid & 31;

    const int dst = edge_index[e];            // row 0 of edge_index
    const int src = edge_index[n_edges + e];  // row 1 of edge_index

    if (lane == 0)
        __hip_atomic_fetch_add(&deg[dst], 1, __ATOMIC_RELAXED, __HIP_MEMORY_SCOPE_AGENT);

    const float4 v = *reinterpret_cast<const float4*>(h + (size_t)src * HID + lane * 4);
    float* o = out + (size_t)dst * HID + lane * 4;
    __hip_atomic_fetch_add(&o[0], v.x, __ATOMIC_RELAXED, __HIP_MEMORY_SCOPE_AGENT);
    __hip_atomic_fetch_add(&o[1], v.y, __ATOMIC_RELAXED, __HIP_MEMORY_SCOPE_AGENT);
    __hip_atomic_fetch_add(&o[2], v.z, __ATOMIC_RELAXED, __HIP_MEMORY_SCOPE_AGENT);
    __hip_atomic_fetch_add(&o[3], v.w, __ATOMIC_RELAXED, __HIP_MEMORY_SCOPE_AGENT);
}

// ---------------------------------------------------------------------------
// Kernel 4: out[n,:] /= (deg[n] + 1), float4 per thread.
// ---------------------------------------------------------------------------
__global__ __launch_bounds__(256)
void gcn_finalize(float* __restrict__ out, const int* __restrict__ deg, int n_nodes)
{
    const int tid   = blockIdx.x * blockDim.x + threadIdx.x;
    const int total = n_nodes * (HID / 4);
    if (tid >= total) return;
    const int node  = tid >> 5;               // 32 float4 per node
    const float s   = 1.0f / (float)(deg[node] + 1);
    float4* p = reinterpret_cast<float4*>(out) + tid;
    float4 v = *p;
    v.x *= s; v.y *= s; v.z *= s; v.w *= s;
    *p = v;
}

// ---------------------------------------------------------------------------
extern "C" void kernel_launch(void* const* d_in, const int* in_sizes, int n_in,
                              void* d_out, int out_size, void* d_ws, size_t ws_size,
                              hipStream_t stream)
{
    const float* x          = (const float*)d_in[0];   // [N, 128]
    const int*   edge_index = (const int*)  d_in[1];   // [2, E] (integer -> int32 per harness)
    const float* W          = (const float*)d_in[2];   // [128, 128]
    const float* b          = (const float*)d_in[3];   // [128]
    float*       out        = (float*)d_out;           // [N, 128]

    const int n_nodes = in_sizes[0] / HID;
    const int n_edges = in_sizes[1] / 2;

    float* h   = (float*)d_ws;                                   // [N,128] = 25.6 MB
    int*   deg = (int*)((char*)d_ws + (size_t)n_nodes * HID * sizeof(float));

    // 1) GEMM: h = x@W^T + b ; out = h
    {
        const int waves  = (n_nodes + 15) / 16;
        const int blocks = (waves + 3) / 4;          // 4 waves / 128-thread block
        gcn_gemm_wmma<<<blocks, 128, 0, stream>>>(x, W, b, h, out, n_nodes);
    }
    // 2) deg = 0
    gcn_zero_deg<<<(n_nodes + 255) / 256, 256, 0, stream>>>(deg, n_nodes);
    // 3) out[dst] += h[src]; deg[dst] += 1
    {
        const long long threads = (long long)n_edges * 32;
        const int blocks = (int)((threads + 255) / 256);
        gcn_scatter<<<blocks, 256, 0, stream>>>(edge_index, h, out, deg, n_edges);
    }
    // 4) out /= (deg + 1)
    {
        const int total = n_nodes * (HID / 4);
        gcn_finalize<<<(total + 255) / 256, 256, 0, stream>>>(out, deg, n_nodes);
    }
}